// UHGAnomalyLoss_12524124635378
// MI455X (gfx1250) — compile-verified
//
#include <hip/hip_runtime.h>

// UHG anomaly loss for MI455X (gfx1250, wave32).
//
// Memory-bound gather-reduce. Strategy:
//  - Compact z[:, :2] into a 1.6 MB (x,y) table in d_ws -> all 25.6M random
//    gathers hit L2 (192 MB), never HBM.
//  - Stream edge_index (102 MB) once with coalesced int4 loads + prefetch.
//  - Deterministic tree reductions (per-block partials, no float atomics) so
//    graph replays are bit-identical.
//  - Wave reduction uses V_WMMA_F32_16X16X4_F32 (A = lane partials, B = ones):
//    one matrix op folds 32 lanes -> 16 row-sums, finished by one xor-shuffle.

#define EPS 1e-9f

static constexpr int kN          = 200000;
static constexpr int kE          = 12800000;
static constexpr int kDim        = 128;
static constexpr int kThreads    = 256;
static constexpr int kEdgeBlocks = 1250;                       // 1250*256 threads * 10 iters * 4 edges = 12.8M exactly
static constexpr int kNodeBlocks = (kN + kThreads - 1) / kThreads; // 782

typedef __attribute__((ext_vector_type(2))) float v2f;
typedef __attribute__((ext_vector_type(8))) float v8f;

__device__ __forceinline__ float safe_div(float num, float den) {
    // Replicates jnp: safe = max(|den|, EPS) * sign(den); sign(0)=0 -> inf/nan like JAX.
    float sgn  = (den > 0.f) ? 1.f : ((den < 0.f) ? -1.f : 0.f);
    float safe = fmaxf(fabsf(den), EPS) * sgn;
    return num / safe;
}

// Wave32 sum reduction. WMMA path: a = {s, 0} puts lane m's value at A[m][0]
// (lanes 0-15) / A[m][2] (lanes 16-31); B = ones 4x16 gives D[m][n] = s_m + s_{m+16}.
// Summing D's 8 VGPRs yields the half-wave sums; one xor-16 shuffle completes it.
__device__ __forceinline__ float wave_reduce(float s) {
#if __has_builtin(__builtin_amdgcn_wmma_f32_16x16x4_f32)
    v2f a; a[0] = s;   a[1] = 0.f;
    v2f b; b[0] = 1.f; b[1] = 1.f;
    v8f c = {};
    v8f d = __builtin_amdgcn_wmma_f32_16x16x4_f32(false, a, false, b, (short)0, c,
                                                  false, false);
    float p = ((d[0] + d[1]) + (d[2] + d[3])) + ((d[4] + d[5]) + (d[6] + d[7]));
    p += __shfl_xor(p, 16, 32);
    return p;
#else
    for (int off = 16; off > 0; off >>= 1) s += __shfl_xor(s, off, 32);
    return s;
#endif
}

// Full-block deterministic sum; valid in all lanes of wave 0 (use on thread 0).
__device__ __forceinline__ float block_reduce(float v, float* lds) {
    float w   = wave_reduce(v);
    int lane  = threadIdx.x & 31;
    int wid   = threadIdx.x >> 5;
    if (lane == 0) lds[wid] = w;
    __syncthreads();
    float r = 0.f;
    if (wid == 0) {
        int nw = (blockDim.x + 31) >> 5;
        r = (lane < nw) ? lds[lane] : 0.f;
        for (int off = 16; off > 0; off >>= 1) r += __shfl_xor(r, off, 32);
    }
    __syncthreads();
    return r;
}

__device__ __forceinline__ float edge_quad(const float2* __restrict__ xy, int s, int d) {
    float2 ps = xy[s];
    float2 pd = xy[d];
    float aas = 1.f - fmaf(ps.x, ps.x, ps.y * ps.y);     // uhg_inner(a,a)
    float aad = 1.f - fmaf(pd.x, pd.x, pd.y * pd.y);
    float ab  = 1.f - fmaf(ps.x, pd.x, ps.y * pd.y);     // uhg_inner(a,b)
    float den = aas * aad;
    float num = fmaf(ab, ab, -den);
    return safe_div(num, den);
}

// ---- Kernel 1: compact (x,y) table + compactness partial sums -----------------
__global__ __launch_bounds__(kThreads)
void uhg_node_kernel(const float* __restrict__ z, float2* __restrict__ xy,
                     float* __restrict__ npart) {
    __shared__ float lds[kThreads / 32];
    int i = blockIdx.x * blockDim.x + threadIdx.x;
    float q = 0.f;
    if (i < kN) {
        float2 p = *(const float2*)(z + (size_t)i * kDim);   // row stride 512B -> aligned
        xy[i] = p;
        float r2 = fmaf(p.x, p.x, p.y * p.y);
        float aa = 1.f - r2;          // den; num = 1 - aa = r2 (origin quadrance)
        q = safe_div(r2, aa);
    }
    float total = block_reduce(q, lds);   // EXEC all-ones here (no divergence)
    if (threadIdx.x == 0) npart[blockIdx.x] = total;
}

// ---- Kernel 2: proximity over 12.8M edges ------------------------------------
__global__ __launch_bounds__(kThreads)
void uhg_edge_kernel(const int* __restrict__ ei, const float2* __restrict__ xy,
                     float* __restrict__ epart) {
    __shared__ float lds[kThreads / 32];
    const int   nthreads = gridDim.x * blockDim.x;        // 320,000
    const int   gtid     = blockIdx.x * blockDim.x + threadIdx.x;
    const int   Eq       = kE / 4;                        // 3,200,000 int4 groups
    const int4* src4 = (const int4*)ei;                   // row 0: sources
    const int4* dst4 = (const int4*)(ei + kE);            // row 1: destinations

    float sum = 0.f;
    for (int i = gtid; i < Eq; i += nthreads) {           // exactly 10 iters/thread
        if (i + nthreads < Eq) {                          // stream-ahead hints
            __builtin_prefetch(&src4[i + nthreads], 0, 1);
            __builtin_prefetch(&dst4[i + nthreads], 0, 1);
        }
        int4 s = src4[i];
        int4 d = dst4[i];
        sum += edge_quad(xy, s.x, d.x);
        sum += edge_quad(xy, s.y, d.y);
        sum += edge_quad(xy, s.z, d.z);
        sum += edge_quad(xy, s.w, d.w);
    }
    float total = block_reduce(sum, lds);                 // EXEC all-ones (exact divide)
    if (threadIdx.x == 0) epart[blockIdx.x] = total;
}

// ---- Kernel 3: fold partials, spread term, final loss ------------------------
__global__ __launch_bounds__(kThreads)
void uhg_finalize_kernel(const int* __restrict__ ei, const float2* __restrict__ xy,
                         const float* __restrict__ epart, const float* __restrict__ npart,
                         float* __restrict__ out) {
    __shared__ float lds[kThreads / 32];
    float se = 0.f, sn = 0.f;
    for (int i = threadIdx.x; i < kEdgeBlocks; i += kThreads) se += epart[i];
    for (int i = threadIdx.x; i < kNodeBlocks; i += kThreads) sn += npart[i];
    float prox_sum = block_reduce(se, lds);
    float comp_sum = block_reduce(sn, lds);

    if (threadIdx.x == 0) {
        // Spread over first 10 edges on lines = cross((x,y,1),(0,0,1)) = (y,-x,0).
        float sp = 0.f;
        for (int e = 0; e < 10; ++e) {
            int    s  = ei[e];
            int    d  = ei[kE + e];
            float2 ps = xy[s];
            float2 pd = xy[d];
            float aas = -(fmaf(ps.x, ps.x, ps.y * ps.y));
            float aad = -(fmaf(pd.x, pd.x, pd.y * pd.y));
            float ab  = -(fmaf(ps.x, pd.x, ps.y * pd.y));
            float den = aas * aad;
            float num = fmaf(ab, ab, -den);
            sp += safe_div(num, den);
        }
        float loss = prox_sum / (float)kE       // proximity mean
                   + comp_sum / (float)kN      // compactness mean
                   + 0.1f * (sp / 10.f);       // SPREAD_WEIGHT * spread mean
        out[0] = loss;
    }
}

extern "C" void kernel_launch(void* const* d_in, const int* in_sizes, int n_in,
                              void* d_out, int out_size, void* d_ws, size_t ws_size,
                              hipStream_t stream) {
    const float* z  = (const float*)d_in[0];   // [200000, 128] f32
    const int*   ei = (const int*)d_in[1];     // [2, 12800000] i32
    float*       out = (float*)d_out;

    // Workspace: 1.6 MB compact node table, then per-block partials.
    float2* xy    = (float2*)d_ws;
    float*  epart = (float*)((char*)d_ws + sizeof(float2) * (size_t)kN);
    float*  npart = epart + kEdgeBlocks;

    uhg_node_kernel<<<kNodeBlocks, kThreads, 0, stream>>>(z, xy, npart);
    uhg_edge_kernel<<<kEdgeBlocks, kThreads, 0, stream>>>(ei, xy, epart);
    uhg_finalize_kernel<<<1, kThreads, 0, stream>>>(ei, xy, epart, npart, out);
}